// OhemCrossEntropy_76733885710773
// MI455X (gfx1250) — compile-verified
//
#include <hip/hip_runtime.h>
#include <hip/hip_bf16.h>

// OHEM BCE loss: scalar = mean of BCE losses where pred < max(kth_order_stat(pred), 0.7)
// 3-pass radix select on float bits (all preds > 0 => bits are order-isomorphic),
// then one fused masked-BCE reduction pass streamed through LDS with the CDNA5
// Tensor Data Mover (double-buffered, TENSORcnt-tracked). Memory-bound:
// ~640MB total traffic => ~27us floor on MI455X @ 23.3 TB/s.

#if defined(__has_builtin)
#if __has_builtin(__builtin_amdgcn_tensor_load_to_lds) && \
    __has_builtin(__builtin_amdgcn_s_wait_tensorcnt)
#define HAVE_TDM 1
#endif
#endif

#if defined(__has_include)
#if __has_include(<hip/amd_detail/amd_gfx1250_TDM.h>)
#define TDM_6ARG 1  // therock-10.0 headers => clang-23 6-arg builtin
#endif
#endif

namespace {

struct Ctrl {
  double loss_sum;        // byte 0
  unsigned int count;     // byte 8
  unsigned int prefix;    // byte 12  (accumulated high bits of selected value)
  unsigned int rank;      // byte 16  (remaining rank within selected bucket)
  float threshold;        // byte 20
  unsigned int pad[10];   // pad to 64 bytes
};

constexpr int TILE = 4096;  // floats per TDM tile (16KB)

__global__ void init_kernel(unsigned int* ws_words, int total_words, unsigned int k) {
  for (int i = 16 + (int)threadIdx.x; i < total_words; i += (int)blockDim.x) ws_words[i] = 0u;
  if (threadIdx.x == 0) {
    Ctrl* c = (Ctrl*)ws_words;
    c->loss_sum = 0.0;
    c->count = 0u;
    c->prefix = 0u;
    c->rank = k;
    c->threshold = 0.7f;
  }
}

// ---------------- radix-select histogram passes (direct global loads) ----------------
template <int SHIFT, int NBITS, bool FILTER>
__global__ __launch_bounds__(256) void hist_kernel(const float4* __restrict__ p4, long long n4,
                                                   const float* __restrict__ ptail,
                                                   long long tail_begin, long long n,
                                                   unsigned int* __restrict__ hist,
                                                   const Ctrl* __restrict__ c) {
  constexpr int BINS = 1 << NBITS;
  __shared__ unsigned int lh[BINS];
  for (int i = threadIdx.x; i < BINS; i += blockDim.x) lh[i] = 0u;
  const unsigned int pref = FILTER ? c->prefix : 0u;
  __syncthreads();

  const long long stride = (long long)gridDim.x * blockDim.x;
  for (long long i = (long long)blockIdx.x * blockDim.x + threadIdx.x; i < n4; i += stride) {
    if (i + 2 * stride < n4) __builtin_prefetch(p4 + i + 2 * stride, 0, 0);  // global_prefetch_b8
    float4 v = p4[i];
    float a[4] = {v.x, v.y, v.z, v.w};
#pragma unroll
    for (int j = 0; j < 4; ++j) {
      unsigned int u = __float_as_uint(a[j]);
      if (!FILTER || (u >> (SHIFT + NBITS)) == pref)
        atomicAdd(&lh[(u >> SHIFT) & (BINS - 1)], 1u);  // ds_add_u32
    }
  }
  if (blockIdx.x == 0 && threadIdx.x == 0) {
    for (long long i = tail_begin; i < n; ++i) {
      unsigned int u = __float_as_uint(ptail[i]);
      if (!FILTER || (u >> (SHIFT + NBITS)) == pref)
        atomicAdd(&lh[(u >> SHIFT) & (BINS - 1)], 1u);
    }
  }
  __syncthreads();
  for (int i = threadIdx.x; i < BINS; i += blockDim.x)
    if (lh[i]) atomicAdd(&hist[i], lh[i]);
}

__global__ __launch_bounds__(256) void select_kernel(const unsigned int* __restrict__ hist,
                                                     int nbins, int nbits, int is_final,
                                                     Ctrl* __restrict__ c) {
  const int t = threadIdx.x;
  const int G = nbins >> 8;  // bins per thread: 8 (2048 bins) or 2 (512 bins)
  unsigned int vals[8];
  unsigned int local = 0u;
#pragma unroll
  for (int j = 0; j < 8; ++j) {
    unsigned int v = (j < G) ? hist[t * G + j] : 0u;
    vals[j] = v;
    local += v;
  }
  const unsigned int rank = c->rank;  // read before any thread can write it
  __shared__ unsigned int sc[256];
  sc[t] = local;
  __syncthreads();
  for (int off = 1; off < 256; off <<= 1) {  // Hillis-Steele inclusive scan
    unsigned int v = (t >= off) ? sc[t - off] : 0u;
    __syncthreads();
    sc[t] += v;
    __syncthreads();
  }
  const unsigned int incl = sc[t];
  const unsigned int excl = incl - local;
  if (excl <= rank && rank < incl) {  // exactly one winner thread
    unsigned int cum = excl;
    int chosen = 0;
    for (int j = 0; j < G; ++j) {
      if (rank < cum + vals[j]) { chosen = j; break; }
      cum += vals[j];
    }
    const unsigned int pref = (c->prefix << nbits) | (unsigned int)(t * G + chosen);
    c->prefix = pref;
    c->rank = rank - cum;
    if (is_final) {
      // pref is now the exact IEEE bit pattern of p_sorted[k]
      c->threshold = fmaxf(__uint_as_float(pref), 0.7f);
    }
  }
}

// ---------------- masked-BCE reduction ----------------

__device__ __forceinline__ void block_reduce_commit(float sum, unsigned int cnt,
                                                    Ctrl* __restrict__ c) {
  // wave32 reduction (gfx1250 is wave32-only)
#pragma unroll
  for (int off = 16; off > 0; off >>= 1) {
    sum += __shfl_down(sum, off, 32);
    cnt += __shfl_down(cnt, off, 32);
  }
  __shared__ float ssum[8];
  __shared__ unsigned int scnt[8];
  const int lane = threadIdx.x & 31;
  const int wave = threadIdx.x >> 5;
  if (lane == 0) { ssum[wave] = sum; scnt[wave] = cnt; }
  __syncthreads();
  if (wave == 0) {
    const int nwaves = blockDim.x >> 5;
    float s = (lane < nwaves) ? ssum[lane] : 0.0f;
    unsigned int q = (lane < nwaves) ? scnt[lane] : 0u;
#pragma unroll
    for (int off = 4; off > 0; off >>= 1) {
      s += __shfl_down(s, off, 32);
      q += __shfl_down(q, off, 32);
    }
    if (lane == 0) {
      atomicAdd(&c->loss_sum, (double)s);  // global_atomic_add_f64
      atomicAdd(&c->count, q);
    }
  }
}

__device__ __forceinline__ void bce_accum(float pv, float tv, float thr, float& sum,
                                          unsigned int& cnt) {
  if (pv < thr) {
    // target is binary: only one log term survives; clamp mirrors torch's -100 log clamp.
    float x = (tv > 0.5f) ? pv : (1.0f - pv);
    sum += fminf(-__logf(x), 100.0f);
    cnt += 1u;
  }
}

#if defined(HAVE_TDM)
typedef unsigned int u32x4 __attribute__((ext_vector_type(4)));
typedef int i32x8 __attribute__((ext_vector_type(8)));
typedef int i32x4 __attribute__((ext_vector_type(4)));

// Issue a 1-D TDM copy of nelem f32 from global to LDS (one op per wave, EXEC-ignored).
// D# layout per CDNA5 ISA §8 (CDNA5 ISA): group0 = {count=1, lds_addr,
// global_addr[56:0], type=2}; group1 = {data_size=4B, tensor_dim0=tile_dim0=nelem, 1-D}.
__device__ __forceinline__ void tdm_load_1d(const void* gptr, void* lptr, int nelem) {
  unsigned long long ga = (unsigned long long)(uintptr_t)gptr;
  unsigned int la = (unsigned int)(uintptr_t)lptr;  // low 32 bits = LDS byte offset
  u32x4 g0;
  g0[0] = 1u;                                                    // count=1, user-mode
  g0[1] = la;                                                    // lds_addr
  g0[2] = (unsigned int)(ga & 0xffffffffull);                    // global_addr[31:0]
  g0[3] = (unsigned int)((ga >> 32) & 0x01ffffffull) | 0x80000000u;  // [56:32] | type=2
  i32x8 g1;
  g1[0] = 0x20000;                                       // data_size=2 (4 bytes), no mcast
  g1[1] = (int)(((unsigned)nelem & 0xffffu) << 16);      // tensor_dim0[15:0]  (bits 63:48)
  g1[2] = (int)((unsigned)nelem >> 16);                  // tensor_dim0[31:16] (bits 79:64)
  g1[3] = (int)(((unsigned)nelem & 0xffffu) << 16);      // tile_dim0 (bits 127:112)
  g1[4] = 0;                                             // tile_dim1=0, tile_dim2=0 (1-D)
  g1[5] = nelem;                                         // tensor_dim0_stride[31:0]
  g1[6] = 0;
  g1[7] = 0;
  i32x4 z4 = {0, 0, 0, 0};
#if defined(TDM_6ARG)
  i32x8 z8 = {0, 0, 0, 0, 0, 0, 0, 0};
  __builtin_amdgcn_tensor_load_to_lds(g0, g1, z4, z4, z8, 0);
#else
  __builtin_amdgcn_tensor_load_to_lds(g0, g1, z4, z4, 0);
#endif
}

// Double-buffered TDM streaming: wave 0 DMAs tile t+1 while all 8 waves consume tile t
// from LDS. TENSORcnt ops complete in order per wave, so wait(2) releases tile t while
// tile t+1's two loads stay in flight.
__global__ __launch_bounds__(256) void loss_kernel_tdm(const float* __restrict__ pred,
                                                       const float* __restrict__ targ,
                                                       long long n, Ctrl* __restrict__ c) {
  __shared__ float sp[2][TILE];
  __shared__ float st[2][TILE];
  const float thr = c->threshold;  // uniform
  float sum = 0.0f;
  unsigned int cnt = 0u;
  const long long ntiles = n / TILE;
  const int wave = threadIdx.x >> 5;
  const long long t0 = blockIdx.x;

  if (wave == 0 && t0 < ntiles) {  // prologue: DMA first tile into buffer 0
    tdm_load_1d(pred + t0 * TILE, &sp[0][0], TILE);
    tdm_load_1d(targ + t0 * TILE, &st[0][0], TILE);
  }
  int b = 0;
  for (long long t = t0; t < ntiles; t += (long long)gridDim.x, b ^= 1) {
    const long long tn = t + (long long)gridDim.x;
    if (wave == 0) {
      if (tn < ntiles) {  // buffer b^1 was drained before last iteration's barrier
        tdm_load_1d(pred + tn * TILE, &sp[b ^ 1][0], TILE);
        tdm_load_1d(targ + tn * TILE, &st[b ^ 1][0], TILE);
        __builtin_amdgcn_s_wait_tensorcnt(2);  // tile t landed; tile t+1 in flight
      } else {
        __builtin_amdgcn_s_wait_tensorcnt(0);
      }
    }
    __syncthreads();  // tile t visible to all waves
#pragma unroll
    for (int j = 0; j < TILE / 256; ++j) {
      const int idx = (int)threadIdx.x + j * 256;  // stride-256 f32: conflict-free
      bce_accum(sp[b][idx], st[b][idx], thr, sum, cnt);
    }
    __syncthreads();  // everyone done reading buffer b before it is re-filled
  }
  if (blockIdx.x == 0) {  // remainder (n % TILE) via direct loads
    for (long long i = ntiles * TILE + threadIdx.x; i < n; i += blockDim.x)
      bce_accum(pred[i], targ[i], thr, sum, cnt);
  }
  block_reduce_commit(sum, cnt, c);
}
#endif  // HAVE_TDM

// Fallback: direct float4 streaming loss pass (used only if TDM builtins are absent).
__global__ __launch_bounds__(256) void loss_kernel_direct(const float4* __restrict__ p4,
                                                          const float4* __restrict__ t4,
                                                          long long n4,
                                                          const float* __restrict__ ptail,
                                                          const float* __restrict__ ttail,
                                                          long long tail_begin, long long n,
                                                          Ctrl* __restrict__ c) {
  const float thr = c->threshold;
  float sum = 0.0f;
  unsigned int cnt = 0u;
  const long long stride = (long long)gridDim.x * blockDim.x;
  for (long long i = (long long)blockIdx.x * blockDim.x + threadIdx.x; i < n4; i += stride) {
    if (i + 2 * stride < n4) {
      __builtin_prefetch(p4 + i + 2 * stride, 0, 0);
      __builtin_prefetch(t4 + i + 2 * stride, 0, 0);
    }
    float4 p = p4[i];
    float t = 0.0f;
    float4 tv = t4[i];
    float pa[4] = {p.x, p.y, p.z, p.w};
    float ta[4] = {tv.x, tv.y, tv.z, tv.w};
    (void)t;
#pragma unroll
    for (int j = 0; j < 4; ++j) bce_accum(pa[j], ta[j], thr, sum, cnt);
  }
  if (blockIdx.x == 0 && threadIdx.x == 0) {
    for (long long i = tail_begin; i < n; ++i) bce_accum(ptail[i], ttail[i], thr, sum, cnt);
  }
  block_reduce_commit(sum, cnt, c);
}

__global__ void finalize_kernel(const Ctrl* __restrict__ c, float* __restrict__ out) {
  if (threadIdx.x == 0 && blockIdx.x == 0)
    out[0] = (float)(c->loss_sum / (double)c->count);
}

}  // namespace

extern "C" void kernel_launch(void* const* d_in, const int* in_sizes, int n_in,
                              void* d_out, int out_size, void* d_ws, size_t ws_size,
                              hipStream_t stream) {
  const float* pred = (const float*)d_in[0];
  const float* targ = (const float*)d_in[1];
  float* out = (float*)d_out;

  const long long n = (long long)in_sizes[0];
  const long long n4 = n >> 2;
  const long long tail_begin = n4 << 2;
  long long kk = (n - 1) / 2;  // min(int(0.5*(n-1)), n-1)
  if (kk > n - 1) kk = n - 1;
  const unsigned int k = (unsigned int)kk;

  unsigned int* ws = (unsigned int*)d_ws;
  Ctrl* c = (Ctrl*)d_ws;
  unsigned int* hist1 = ws + 16;       // 2048 bins: float bits [30:20]
  unsigned int* hist2 = hist1 + 2048;  // 2048 bins: bits [19:9]
  unsigned int* hist3 = hist2 + 2048;  //  512 bins: bits [8:0]
  const int total_words = 16 + 2048 + 2048 + 512;  // 18,496 bytes of ws used

  const int blocks = 2048;
  const int threads = 256;  // 8 wave32s per block

  init_kernel<<<1, 1024, 0, stream>>>(ws, total_words, k);

  hist_kernel<20, 11, false><<<blocks, threads, 0, stream>>>(
      (const float4*)pred, n4, pred, tail_begin, n, hist1, c);
  select_kernel<<<1, 256, 0, stream>>>(hist1, 2048, 11, 0, c);

  hist_kernel<9, 11, true><<<blocks, threads, 0, stream>>>(
      (const float4*)pred, n4, pred, tail_begin, n, hist2, c);
  select_kernel<<<1, 256, 0, stream>>>(hist2, 2048, 11, 0, c);

  hist_kernel<0, 9, true><<<blocks, threads, 0, stream>>>(
      (const float4*)pred, n4, pred, tail_begin, n, hist3, c);
  select_kernel<<<1, 256, 0, stream>>>(hist3, 512, 9, 1, c);

#if defined(HAVE_TDM)
  loss_kernel_tdm<<<blocks, threads, 0, stream>>>(pred, targ, n, c);
#else
  loss_kernel_direct<<<blocks, threads, 0, stream>>>(
      (const float4*)pred, (const float4*)targ, n4, pred, targ, tail_begin, n, c);
#endif

  finalize_kernel<<<1, 1, 0, stream>>>(c, out);
}